// CRF_62904091018078
// MI455X (gfx1250) — compile-verified
//
#include <hip/hip_runtime.h>
#include <math.h>

typedef __attribute__((ext_vector_type(2))) float v2f;
typedef __attribute__((ext_vector_type(8))) float v8f;

#define CRF_B 256
#define CRF_T 1024
#define CRF_L 128
#define BT 16    // batches per block (WMMA M)
#define ST 132   // padded LDS row stride in floats (avoids bank conflicts)

// fast native transcendentals: v_exp_f32 / v_log_f32 (base-2 HW ops)
__device__ __forceinline__ float fast_exp(float x) {
    return __builtin_amdgcn_exp2f(x * 1.44269504088896340736f);   // exp(x)=2^(x*log2 e)
}
__device__ __forceinline__ float fast_log(float x) {
    return __builtin_amdgcn_logf(x) * 0.69314718055994530942f;    // ln(x)=log2(x)*ln2
}

// ---------------------------------------------------------------------------
// Forward recursion: 16 blocks x 256 threads (8 waves). Each block = 16 batch
// rows; each wave = one 16-wide column tile. Per timestep each wave issues
// 32 x v_wmma_f32_16x16x4_f32 with the constant exp(trans) B-operand held in
// 64 VGPRs for the whole T loop.
// ---------------------------------------------------------------------------
__global__ __launch_bounds__(256) void crf_forward(
    const float* __restrict__ h, const float* __restrict__ mask,
    const float* __restrict__ trans, const float* __restrict__ start_trans,
    const float* __restrict__ end_trans, float* __restrict__ den)
{
    __shared__ float s_score[BT * ST];
    __shared__ float s_P[BT * ST];
    __shared__ float s_rowmax[BT];
    __shared__ float s_mask[BT];

    const int tid  = threadIdx.x;
    const int b0   = blockIdx.x * BT;
    const int wave = tid >> 5;     // 0..7 : column tile (j = wave*16 .. +15)
    const int lane = tid & 31;
    const int hf   = lane >> 4;    // half-wave select
    const int l16  = lane & 15;
    const int row  = tid >> 4;     // 0..15 : batch row for reduction phases
    const int sub  = tid & 15;

    // ---- preload B operand E = exp(trans) in WMMA 4x16 f32 B layout ----
    // VGPR0: lanes0-15 K=4kk+0, lanes16-31 K=4kk+2 ; VGPR1: K=4kk+1 / 4kk+3
    v2f eb[32];
    {
        const int col = wave * 16 + l16;
#pragma unroll
        for (int kk = 0; kk < 32; ++kk) {
            const int k0 = kk * 4 + hf * 2;
            v2f e;
            e.x = expf(trans[k0 * CRF_L + col]);
            e.y = expf(trans[(k0 + 1) * CRF_L + col]);
            eb[kk] = e;
        }
    }

    // ---- t = 0 : score = start_trans + h[:,0,:] ----
    {
        const float* hrow = h + (size_t)(b0 + row) * CRF_T * CRF_L;
#pragma unroll
        for (int k = 0; k < 8; ++k) {
            const int j = sub + 16 * k;
            s_score[row * ST + j] = start_trans[j] + hrow[j];
        }
    }
    __syncthreads();

    // ---- sequential forward recurrence over T ----
    for (int t = 1; t < CRF_T; ++t) {
        // phase A: per-row max + P = exp(score - max); stage mask value
        float sv[8];
        float mx = -1e30f;
#pragma unroll
        for (int k = 0; k < 8; ++k) {
            sv[k] = s_score[row * ST + sub + 16 * k];
            mx = fmaxf(mx, sv[k]);
        }
#pragma unroll
        for (int m = 1; m < 16; m <<= 1)
            mx = fmaxf(mx, __shfl_xor(mx, m, 32));
        if (sub == 0) s_rowmax[row] = mx;
        if (sub == 1) s_mask[row] = mask[(size_t)(b0 + row) * CRF_T + t];
#pragma unroll
        for (int k = 0; k < 8; ++k)
            s_P[row * ST + sub + 16 * k] = fast_exp(sv[k] - mx);
        __syncthreads();

        // phase B: C(16x16) = P(16x128) x E(128x16 tile), f32 WMMA chain
        v8f c = {};
        const float* Pbase = &s_P[l16 * ST + hf * 2];   // A: M=l16, K base per half
#pragma unroll
        for (int kk = 0; kk < 32; ++kk) {
            v2f a = *(const v2f*)(Pbase + kk * 4);       // ds_load_b64
            c = __builtin_amdgcn_wmma_f32_16x16x4_f32(
                    false, a, false, eb[kk], (short)0, c, false, false);
        }

        // phase C: score_new = h_t + rowmax + log(c), masked blend, write back
        const int j = wave * 16 + l16;
#pragma unroll
        for (int r = 0; r < 8; ++r) {
            const int gm   = r + hf * 8;                 // C/D layout: M=r / r+8
            const float oldv = s_score[gm * ST + j];
            const float hv   = h[(size_t)(b0 + gm) * CRF_T * CRF_L
                                 + (size_t)t * CRF_L + j];
            const float sval = hv + s_rowmax[gm] + fast_log(c[r]);
            const float mval = s_mask[gm];
            s_score[gm * ST + j] = sval * mval + oldv * (1.0f - mval);
        }
        __syncthreads();
    }

    // ---- den = logsumexp(score_T + end_trans) per batch row ----
    {
        float vv[8];
        float mx = -1e30f;
#pragma unroll
        for (int k = 0; k < 8; ++k) {
            const int j = sub + 16 * k;
            vv[k] = s_score[row * ST + j] + end_trans[j];
            mx = fmaxf(mx, vv[k]);
        }
#pragma unroll
        for (int m = 1; m < 16; m <<= 1)
            mx = fmaxf(mx, __shfl_xor(mx, m, 32));
        float se = 0.0f;
#pragma unroll
        for (int k = 0; k < 8; ++k) se += fast_exp(vv[k] - mx);
#pragma unroll
        for (int m = 1; m < 16; m <<= 1)
            se += __shfl_xor(se, m, 32);
        if (sub == 0) den[b0 + row] = mx + fast_log(se);
    }
}

// ---------------------------------------------------------------------------
// Numerator: one wave per batch; lane-strided sum over T with shuffle reduce.
// Reads den from workspace (forward kernel ordered earlier on same stream).
// ---------------------------------------------------------------------------
__global__ __launch_bounds__(256) void crf_num(
    const float* __restrict__ h, const int* __restrict__ labels,
    const float* __restrict__ mask, const float* __restrict__ trans,
    const float* __restrict__ start_trans, const float* __restrict__ end_trans,
    const float* __restrict__ den, float* __restrict__ out)
{
    const int b    = blockIdx.x * 8 + (threadIdx.x >> 5);
    const int lane = threadIdx.x & 31;
    const int*   lab  = labels + (size_t)b * CRF_T;
    const float* mrow = mask + (size_t)b * CRF_T;
    const float* hrow = h + (size_t)b * CRF_T * CRF_L;

    float acc = 0.0f, msum = 0.0f;
    for (int t = lane; t < CRF_T; t += 32) {
        const int   lt = lab[t];
        const float mt = mrow[t];
        msum += mt;
        if (t < CRF_T - 1) {
            acc += hrow[(size_t)t * CRF_L + lt] * mt;
            acc += trans[lt * CRF_L + lab[t + 1]] * mrow[t + 1];
        }
    }
#pragma unroll
    for (int m = 1; m < 32; m <<= 1) {
        acc  += __shfl_xor(acc, m, 32);
        msum += __shfl_xor(msum, m, 32);
    }
    if (lane == 0) {
        const int last_idx = (int)msum - 1;
        const int last     = lab[last_idx];
        float num = acc + start_trans[lab[0]]
                  + hrow[(size_t)(CRF_T - 1) * CRF_L + last] * mrow[CRF_T - 1]
                  + end_trans[last];
        out[b] = num - den[b];
    }
}

extern "C" void kernel_launch(void* const* d_in, const int* in_sizes, int n_in,
                              void* d_out, int out_size, void* d_ws, size_t ws_size,
                              hipStream_t stream) {
    const float* h           = (const float*)d_in[0];
    const int*   labels      = (const int*)d_in[1];
    const float* mask        = (const float*)d_in[2];
    const float* trans       = (const float*)d_in[3];
    const float* start_trans = (const float*)d_in[4];
    const float* end_trans   = (const float*)d_in[5];
    float* out = (float*)d_out;
    float* den = (float*)d_ws;   // B floats of scratch

    crf_forward<<<CRF_B / BT, 256, 0, stream>>>(h, mask, trans, start_trans,
                                                end_trans, den);
    crf_num<<<CRF_B / 8, 256, 0, stream>>>(h, labels, mask, trans, start_trans,
                                           end_trans, den, out);
}